// AttnDecoderRNN_22247930593355
// MI455X (gfx1250) — compile-verified
//
#include <hip/hip_runtime.h>
#include <math.h>

// Problem constants (match reference)
#define H 2048
#define V 50257
#define L 20

typedef __attribute__((ext_vector_type(2))) float v2f;
typedef __attribute__((ext_vector_type(8))) float v8f;

// Async-to-LDS builtin operand types: pointer to 4 x i32, AS1 / AS3
typedef int v4i_t __attribute__((vector_size(16)));
typedef __attribute__((address_space(1))) v4i_t GlobV4;
typedef __attribute__((address_space(3))) v4i_t LdsV4;

// Workspace layout (float offsets into d_ws)
#define WS_XCAT2 0            // [embedded, context]  : 2H
#define WS_X     (2 * H)      // relu(comb)           : H
#define WS_GI    (3 * H)      // w_ih @ x + b_ih      : 3H
#define WS_GH    (6 * H)      // w_hh @ h0 + b_hh     : 3H
#define WS_HNEW  (9 * H)      // new hidden           : H

#define LDS_STRIDE 68         // 64 cols + pad 4 (stride % 64 == 4 -> conflict-free b64 reads)
#define BUF_FLOATS (128 * LDS_STRIDE)

// ---------------------------------------------------------------------------
// Kernel 1: embedding gather + attention logits + softmax + context vector.
// ---------------------------------------------------------------------------
__global__ __launch_bounds__(256) void attn_ctx_kernel(
    const int* __restrict__ ids, const float* __restrict__ hidden,
    const float* __restrict__ enc, const float* __restrict__ emb,
    const float* __restrict__ attn_w, const float* __restrict__ attn_b,
    float* __restrict__ ws, float* __restrict__ out_attn)
{
  __shared__ float red[256];
  __shared__ float wts[L];
  const int tid = threadIdx.x;
  const int id = ids[0];
  const float* erow = emb + (size_t)id * H;

  for (int l = 0; l < L; ++l) {
    const float* wrow = attn_w + l * (2 * H);
    float p = 0.f;
    for (int j = tid; j < 2 * H; j += 256) {
      float cv = (j < H) ? erow[j] : hidden[j - H];
      p += cv * wrow[j];
    }
    red[tid] = p;
    __syncthreads();
    for (int s = 128; s > 0; s >>= 1) {
      if (tid < s) red[tid] += red[tid + s];
      __syncthreads();
    }
    if (tid == 0) wts[l] = red[0] + attn_b[l];
    __syncthreads();
  }

  if (tid == 0) {
    float m = wts[0];
    for (int l = 1; l < L; ++l) m = fmaxf(m, wts[l]);
    float s = 0.f;
    for (int l = 0; l < L; ++l) { float e = expf(wts[l] - m); wts[l] = e; s += e; }
    float inv = 1.f / s;
    for (int l = 0; l < L; ++l) { wts[l] *= inv; out_attn[l] = wts[l]; }
  }
  __syncthreads();

  for (int j = tid; j < H; j += 256) {
    float ctx = 0.f;
    #pragma unroll
    for (int l = 0; l < L; ++l) ctx += wts[l] * enc[l * H + j];
    ws[WS_XCAT2 + j]     = erow[j];
    ws[WS_XCAT2 + H + j] = ctx;
  }
}

// ---------------------------------------------------------------------------
// GEMV y = act(A @ x + bias), A row-major [rows,K], K % 64 == 0.
// 256 threads = 8 wave32s; 128 rows/block (16 per wave).
// Weight tiles stream global->LDS via GLOBAL_LOAD_ASYNC_TO_LDS_B128 with
// double buffering (ASYNCcnt pipelining, two barriers per chunk); each wave
// feeds V_WMMA_F32_16X16X4_F32 with A-fragments from LDS (conflict-free,
// stride 68) and B = x in column 0, so D[:,0] is the GEMV result.
// ---------------------------------------------------------------------------
__global__ __launch_bounds__(256) void gemv_wmma_kernel(
    const float* __restrict__ A, const float* __restrict__ xv,
    const float* __restrict__ bias, float* __restrict__ y,
    int rows, int K, int relu)
{
  __shared__ float xs[4096];                 // whole x vector (K <= 4096)
  __shared__ float lds_a[2][BUF_FLOATS];     // double-buffered 128x64 tile

  const int tid  = threadIdx.x;
  const int lane = tid & 31;                 // wave32
  const int wv   = tid >> 5;                 // wave id 0..7
  const int row0 = blockIdx.x * 128;

  for (int j = tid; j < K; j += 256) xs[j] = xv[j];

  // Staging coordinates: thread t stages rows rL+16i, cols cL..cL+3.
  const int rL = tid >> 4;                   // 0..15
  const int cL = (tid & 15) << 2;            // 0..60 step 4

  // Per-chunk issue of 8 async b128 loads (one per 16-row group).
  // OOB rows are clamped into range: garbage values are computed but never
  // stored (row m of D depends only on row m of A).
  auto stage = [&](int k0, int buf) {
    #pragma unroll
    for (int i = 0; i < 8; ++i) {
      int gr = row0 + rL + (i << 4);
      if (gr >= rows) gr = rows - 1;
      const float* src = A + (size_t)gr * K + k0 + cL;
      float* dst = &lds_a[buf][(rL + (i << 4)) * LDS_STRIDE + cL];
      __builtin_amdgcn_global_load_async_to_lds_b128(
          (GlobV4*)src, (LdsV4*)dst, /*offset=*/0, /*cpol=*/0);
    }
  };

  // WMMA fragment addressing: row = wv*16 + (lane&15), col offset 2*(lane>=16)
  const float* ap0 = &lds_a[0][((wv << 4) + (lane & 15)) * LDS_STRIDE + ((lane >> 4) << 1)];
  const bool lead = (lane & 15) == 0;        // lanes 0 and 16 carry B column 0

  v8f c = {};
  const int nchunks = K >> 6;

  stage(0, 0);
  for (int ch = 0; ch < nchunks; ++ch) {
    const int buf = ch & 1;
    const int k0  = ch << 6;
    if (ch + 1 < nchunks) {
      stage(k0 + 64, buf ^ 1);
      __builtin_amdgcn_s_wait_asynccnt(8);   // current buffer's 8 loads landed
    } else {
      __builtin_amdgcn_s_wait_asynccnt(0);
    }
    __syncthreads();                         // all waves' tile data visible

    const float* ap = ap0 + buf * BUF_FLOATS;
    const float* xp = &xs[k0 + ((lane >> 4) << 1)];
    #pragma unroll
    for (int kc = 0; kc < 64; kc += 4) {
      v2f a = *reinterpret_cast<const v2f*>(ap + kc);
      // broadcast ds_load_b64 + cndmask: no EXEC divergence around WMMA
      float2 xl = *reinterpret_cast<const float2*>(xp + kc);
      v2f b;
      b.x = lead ? xl.x : 0.f;
      b.y = lead ? xl.y : 0.f;
      c = __builtin_amdgcn_wmma_f32_16x16x4_f32(
          /*neg_a=*/false, a, /*neg_b=*/false, b,
          /*c_mod=*/(short)0, c, /*reuse_a=*/false, /*reuse_b=*/false);
    }
    __syncthreads();                         // reads done before buf is re-staged
  }

  // D column 0: lane 0 -> rows m=0..7 in c[0..7]; lane 16 -> rows m=8..15.
  if (lane == 0 || lane == 16) {
    const int rbase = row0 + (wv << 4) + ((lane >> 4) << 3);
    #pragma unroll
    for (int m = 0; m < 8; ++m) {
      const int r = rbase + m;
      if (r < rows) {
        float vv = c[m] + bias[r];
        if (relu) vv = fmaxf(vv, 0.f);
        y[r] = vv;
      }
    }
  }
}

// ---------------------------------------------------------------------------
// GRU cell gates (PyTorch order r,z,n) -> h_new
// ---------------------------------------------------------------------------
__global__ __launch_bounds__(256) void gru_gate_kernel(
    const float* __restrict__ gi, const float* __restrict__ gh,
    const float* __restrict__ h0, float* __restrict__ hws,
    float* __restrict__ hout)
{
  const int j = blockIdx.x * 256 + threadIdx.x;
  if (j >= H) return;
  const float r = 1.f / (1.f + expf(-(gi[j] + gh[j])));
  const float z = 1.f / (1.f + expf(-(gi[H + j] + gh[H + j])));
  const float n = tanhf(gi[2 * H + j] + r * gh[2 * H + j]);
  const float h = (1.f - z) * n + z * h0[j];
  hws[j]  = h;
  hout[j] = h;
}

// ---------------------------------------------------------------------------
extern "C" void kernel_launch(void* const* d_in, const int* in_sizes, int n_in,
                              void* d_out, int out_size, void* d_ws, size_t ws_size,
                              hipStream_t stream)
{
  const int*   ids    = (const int*)  d_in[0];
  const float* hidden = (const float*)d_in[1];   // [1,1,H]
  const float* enc    = (const float*)d_in[2];   // [L,H]
  const float* emb    = (const float*)d_in[3];   // [V,H]
  const float* attn_w = (const float*)d_in[4];   // [L,2H]
  const float* attn_b = (const float*)d_in[5];   // [L]
  const float* comb_w = (const float*)d_in[6];   // [H,2H]
  const float* comb_b = (const float*)d_in[7];   // [H]
  const float* w_ih   = (const float*)d_in[8];   // [3H,H]
  const float* w_hh   = (const float*)d_in[9];   // [3H,H]
  const float* b_ih   = (const float*)d_in[10];  // [3H]
  const float* b_hh   = (const float*)d_in[11];  // [3H]
  const float* out_w  = (const float*)d_in[12];  // [V,H]
  const float* out_b  = (const float*)d_in[13];  // [V]
  float* out = (float*)d_out;  // [V logits][H h_new][L attn_weights]
  float* ws  = (float*)d_ws;

  attn_ctx_kernel<<<1, 256, 0, stream>>>(ids, hidden, enc, emb, attn_w, attn_b,
                                         ws, out + V + H);
  gemv_wmma_kernel<<<H / 128, 256, 0, stream>>>(
      comb_w, ws + WS_XCAT2, comb_b, ws + WS_X, H, 2 * H, 1);
  gemv_wmma_kernel<<<(3 * H) / 128, 256, 0, stream>>>(
      w_ih, ws + WS_X, b_ih, ws + WS_GI, 3 * H, H, 0);
  gemv_wmma_kernel<<<(3 * H) / 128, 256, 0, stream>>>(
      w_hh, hidden, b_hh, ws + WS_GH, 3 * H, H, 0);
  gru_gate_kernel<<<(H + 255) / 256, 256, 0, stream>>>(
      ws + WS_GI, ws + WS_GH, hidden, ws + WS_HNEW, out + V);
  gemv_wmma_kernel<<<(V + 127) / 128, 256, 0, stream>>>(
      out_w, ws + WS_HNEW, out_b, out, V, H, 0);
}